// WeightedProcrustes_40785009442991
// MI455X (gfx1250) — compile-verified
//
#include <hip/hip_runtime.h>
#include <math.h>

typedef float fv2 __attribute__((ext_vector_type(2)));
typedef float fv8 __attribute__((ext_vector_type(8)));

#define NPTS 1024

// One 16x16x4 f32 WMMA consumes 16 points: 4 row/col groups (g = m/4) of 4
// points each (k = 0..3).  A[m][k] = p_{m%4}(pt),  B[k][n] = q_{n%4}(pt),
// pt = n0 + 4*(m/4) + k.  Diagonal 4x4 blocks of D accumulate G = sum p q^T.
__device__ __forceinline__ fv8 chunk_wmma(const float* __restrict__ sb,
                                          const float* __restrict__ tb,
                                          const float* __restrict__ wb,
                                          int n0, int e0, int ci, bool isOne,
                                          fv8 acc) {
  int p0 = n0 + e0;           // point for reg 0 (k = 2*half)
  int p1 = p0 + 1;            // point for reg 1 (k = 2*half+1)
  float s0 = sb[p0 * 3 + ci];
  float s1 = sb[p1 * 3 + ci];
  float t0 = tb[p0 * 3 + ci];
  float t1 = tb[p1 * 3 + ci];
  float w0 = wb[p0];
  float w1 = wb[p1];
  w0 = (w0 < 0.5f) ? 0.0f : w0;          // WEIGHT_THRESH
  w1 = (w1 < 0.5f) ? 0.0f : w1;
  fv2 A, Bv;
  A.x  = isOne ? 1.0f : s0;              // p = (sx,sy,sz,1)
  A.y  = isOne ? 1.0f : s1;
  Bv.x = isOne ? w0 : w0 * t0;           // q = (w*tx,w*ty,w*tz,w)
  Bv.y = isOne ? w1 : w1 * t1;
  return __builtin_amdgcn_wmma_f32_16x16x4_f32(false, A, false, Bv,
                                               (short)0, acc, false, false);
}

__global__ __launch_bounds__(256) void procrustes_kernel(
    const float* __restrict__ src, const float* __restrict__ tgt,
    const float* __restrict__ wgt, float* __restrict__ out, int Bn) {
  __shared__ float red[8][256];          // per-wave accumulator dump
  const int lane = threadIdx.x & 31;
  const int wave = threadIdx.x >> 5;
  const int b = blockIdx.x * 8 + wave;   // one wave32 per batch
  const bool wr = (b < Bn);
  const int bc = wr ? b : 0;

  const float* sb = src + (size_t)bc * (NPTS * 3);
  const float* tb = tgt + (size_t)bc * (NPTS * 3);
  const float* wb = wgt + (size_t)bc * NPTS;

  // WMMA fragment coordinates for this lane (A and B want the same points)
  const int half = lane >> 4;            // k-block: lanes 0-15 k={0,1}, 16-31 k={2,3}
  const int m    = lane & 15;            // A row / B col
  const int i    = m & 3;                // vector component (0..2 = xyz, 3 = w-slot)
  const int g    = m >> 2;               // 4-point group
  const int e0   = 4 * g + 2 * half;     // within-chunk point for reg 0
  const int ci   = (i < 3) ? i : 0;      // clamped component for the load
  const bool isOne = (i == 3);

  fv8 acc0 = {0.f, 0.f, 0.f, 0.f, 0.f, 0.f, 0.f, 0.f};
  fv8 acc1 = acc0;
#pragma unroll 2
  for (int n0 = 0; n0 < NPTS; n0 += 32) {   // two independent WMMA chains
    acc0 = chunk_wmma(sb, tb, wb, n0,      e0, ci, isOne, acc0);
    acc1 = chunk_wmma(sb, tb, wb, n0 + 16, e0, ci, isOne, acc1);
  }
  acc0 = acc0 + acc1;

  // D[m][n] -> reg m%8, lane (m/8)*16 + n.  Dump and sum diagonal 4x4 blocks.
  float* myred = &red[wave][0];
#pragma unroll
  for (int r = 0; r < 8; ++r) myred[lane * 8 + r] = acc0[r];
  __syncthreads();

  float Gv = 0.0f;
  {
    int ii = lane >> 2, jj = lane & 3;      // lanes 0..15 own G[ii][jj]
    if (lane < 16) {
      Gv = myred[jj * 8 + ii]                      // g=0: m=ii,      n=jj
         + myred[(4 + jj) * 8 + (4 + ii)]         // g=1: m=4+ii,    n=4+jj
         + myred[(24 + jj) * 8 + ii]              // g=2: m=8+ii,    n=8+jj
         + myred[(28 + jj) * 8 + (4 + ii)];       // g=3: m=12+ii,   n=12+jj
    }
  }
  float G[16];
#pragma unroll
  for (int q = 0; q < 16; ++q) G[q] = __shfl(Gv, q, 32);

  if (lane == 0 && wr) {
    // ---- finalize centered covariance H ----
    float Sw = G[15];
    float inv = 1.0f / (Sw + 1e-5f);
    float sc0 = G[3] * inv, sc1 = G[7] * inv, sc2 = G[11] * inv;   // src centroid
    float tc0 = G[12] * inv, tc1 = G[13] * inv, tc2 = G[14] * inv; // tgt centroid
    float f = 2.0f - Sw * inv;
    float h00 = G[0]*inv - f*sc0*tc0, h01 = G[1]*inv - f*sc0*tc1, h02 = G[2]*inv  - f*sc0*tc2;
    float h10 = G[4]*inv - f*sc1*tc0, h11 = G[5]*inv - f*sc1*tc1, h12 = G[6]*inv  - f*sc1*tc2;
    float h20 = G[8]*inv - f*sc2*tc0, h21 = G[9]*inv - f*sc2*tc1, h22 = G[10]*inv - f*sc2*tc2;

    // ---- 3x3 SVD via Jacobi on B = H^T H ----
    float b00 = h00*h00 + h10*h10 + h20*h20;
    float b01 = h00*h01 + h10*h11 + h20*h21;
    float b02 = h00*h02 + h10*h12 + h20*h22;
    float b11 = h01*h01 + h11*h11 + h21*h21;
    float b12 = h01*h02 + h11*h12 + h21*h22;
    float b22 = h02*h02 + h12*h12 + h22*h22;
    float v00 = 1.f, v01 = 0.f, v02 = 0.f;
    float v10 = 0.f, v11 = 1.f, v12 = 0.f;
    float v20 = 0.f, v21 = 0.f, v22 = 1.f;

    auto rot = [](float &app, float &aqq, float &apq, float &arp, float &arq,
                  float &x0p, float &x0q, float &x1p, float &x1q,
                  float &x2p, float &x2q) {
      float a = apq;
      if (fabsf(a) < 1e-30f) return;
      float tau = (aqq - app) / (2.0f * a);
      float tt = copysignf(1.0f, tau) / (fabsf(tau) + sqrtf(1.0f + tau * tau));
      float c = 1.0f / sqrtf(1.0f + tt * tt);
      float s = tt * c;
      app -= tt * a; aqq += tt * a; apq = 0.0f;
      float rp = arp, rq = arq;
      arp = c * rp - s * rq;  arq = s * rp + c * rq;
      float xp, xq;
      xp = x0p; xq = x0q; x0p = c*xp - s*xq; x0q = s*xp + c*xq;
      xp = x1p; xq = x1q; x1p = c*xp - s*xq; x1q = s*xp + c*xq;
      xp = x2p; xq = x2q; x2p = c*xp - s*xq; x2q = s*xp + c*xq;
    };
#pragma unroll
    for (int it = 0; it < 4; ++it) {
      rot(b00, b11, b01, b02, b12, v00, v01, v10, v11, v20, v21);
      rot(b00, b22, b02, b01, b12, v00, v02, v10, v12, v20, v22);
      rot(b11, b22, b12, b01, b02, v01, v02, v11, v12, v21, v22);
    }
    auto cswap = [](float &la, float &lb, float &a0, float &c0,
                    float &a1, float &c1, float &a2, float &c2) {
      if (la < lb) {
        float t;
        t = la; la = lb; lb = t;
        t = a0; a0 = c0; c0 = t;
        t = a1; a1 = c1; c1 = t;
        t = a2; a2 = c2; c2 = t;
      }
    };
    cswap(b00, b11, v00, v01, v10, v11, v20, v21);
    cswap(b00, b22, v00, v02, v10, v12, v20, v22);
    cswap(b11, b22, v01, v02, v11, v12, v21, v22);

    // right-handed V: col2 = col0 x col1  (det V = +1)
    v02 = v10 * v21 - v20 * v11;
    v12 = v20 * v01 - v00 * v21;
    v22 = v00 * v11 - v10 * v01;

    // U columns from H*v_k, Gram-Schmidt, u2 = u0 x u1 (det U = +1)
    float u0x = h00*v00 + h01*v10 + h02*v20;
    float u0y = h10*v00 + h11*v10 + h12*v20;
    float u0z = h20*v00 + h21*v10 + h22*v20;
    float nn0 = u0x*u0x + u0y*u0y + u0z*u0z;
    float rn0 = 1.0f / sqrtf(fmaxf(nn0, 1e-30f));
    u0x *= rn0; u0y *= rn0; u0z *= rn0;

    float u1x = h00*v01 + h01*v11 + h02*v21;
    float u1y = h10*v01 + h11*v11 + h12*v21;
    float u1z = h20*v01 + h21*v11 + h22*v21;
    float d01 = u0x*u1x + u0y*u1y + u0z*u1z;
    u1x -= d01*u0x; u1y -= d01*u0y; u1z -= d01*u0z;
    float nn1 = u1x*u1x + u1y*u1y + u1z*u1z;
    float rn1 = 1.0f / sqrtf(fmaxf(nn1, 1e-30f));
    u1x *= rn1; u1y *= rn1; u1z *= rn1;

    float u2x = u0y*u1z - u0z*u1y;
    float u2y = u0z*u1x - u0x*u1z;
    float u2z = u0x*u1y - u0y*u1x;

    // R = V U^T  (== V diag(1,1,sgn) U^T of the reference)
    float r00 = v00*u0x + v01*u1x + v02*u2x;
    float r01 = v00*u0y + v01*u1y + v02*u2y;
    float r02 = v00*u0z + v01*u1z + v02*u2z;
    float r10 = v10*u0x + v11*u1x + v12*u2x;
    float r11 = v10*u0y + v11*u1y + v12*u2y;
    float r12 = v10*u0z + v11*u1z + v12*u2z;
    float r20 = v20*u0x + v21*u1x + v22*u2x;
    float r21 = v20*u0y + v21*u1y + v22*u2y;
    float r22 = v20*u0z + v21*u1z + v22*u2z;

    float txv = tc0 - (r00*sc0 + r01*sc1 + r02*sc2);
    float tyv = tc1 - (r10*sc0 + r11*sc1 + r12*sc2);
    float tzv = tc2 - (r20*sc0 + r21*sc1 + r22*sc2);

    float* Ro = out + (size_t)bc * 9;
    Ro[0] = r00; Ro[1] = r01; Ro[2] = r02;
    Ro[3] = r10; Ro[4] = r11; Ro[5] = r12;
    Ro[6] = r20; Ro[7] = r21; Ro[8] = r22;
    float* To = out + (size_t)Bn * 9 + (size_t)bc * 3;
    To[0] = txv; To[1] = tyv; To[2] = tzv;
  }
}

extern "C" void kernel_launch(void* const* d_in, const int* in_sizes, int n_in,
                              void* d_out, int out_size, void* d_ws, size_t ws_size,
                              hipStream_t stream) {
  (void)n_in; (void)out_size; (void)d_ws; (void)ws_size;
  const float* src = (const float*)d_in[0];
  const float* tgt = (const float*)d_in[1];
  const float* wgt = (const float*)d_in[2];
  float* out = (float*)d_out;
  int Bn = in_sizes[2] / NPTS;          // weights = (B, N)
  int blocks = (Bn + 7) / 8;            // 8 waves (batches) per 256-thread block
  hipLaunchKernelGGL(procrustes_kernel, dim3(blocks), dim3(256), 0, stream,
                     src, tgt, wgt, out, Bn);
}